// CtcLoss_21869973471797
// MI455X (gfx1250) — compile-verified
//
#include <hip/hip_runtime.h>
#include <cstdint>

// Problem constants (from reference): T=2000, B=32, V=1000, S=200
#define TT 2000
#define BB 32
#define VV 1000
#define SS 200
#define LL (2 * SS + 1)   // 401 extended states
#define NEGF (-1e30f)
#define NTB 512           // threads for ctc_forward kernel (16 waves)
#define NV4 (VV / 4)      // 250 float4 chunks per row

// ---------------------------------------------------------------------------
// CDNA5 async global->LDS path (guarded; falls back to sync copies)
// ---------------------------------------------------------------------------
#if defined(__has_builtin)
#if __has_builtin(__builtin_amdgcn_global_load_async_to_lds_b32) && \
    __has_builtin(__builtin_amdgcn_s_wait_asynccnt)
#define HAVE_ASYNC_LDS 1
#endif
#if __has_builtin(__builtin_amdgcn_global_load_async_to_lds_b128)
#define HAVE_ASYNC_B128 1
#endif
#endif
#ifndef HAVE_ASYNC_LDS
#define HAVE_ASYNC_LDS 0
#endif
#ifndef HAVE_ASYNC_B128
#define HAVE_ASYNC_B128 0
#endif

// async instructions issued per row by each participating wave
#if HAVE_ASYNC_B128
#define ROW_CNT 1
#else
#define ROW_CNT 4
#endif

#if HAVE_ASYNC_LDS
typedef int v4i __attribute__((__vector_size__(16)));
typedef __attribute__((address_space(1))) int* gas_ptr;    // global scalar
typedef __attribute__((address_space(3))) int* las_ptr;    // LDS scalar
typedef __attribute__((address_space(1))) v4i* gas4_ptr;   // global v4i
typedef __attribute__((address_space(3))) v4i* las4_ptr;   // LDS v4i
__device__ __forceinline__ gas_ptr to_gas(const void* p) {
  return reinterpret_cast<gas_ptr>(reinterpret_cast<uintptr_t>(p));
}
__device__ __forceinline__ las_ptr to_las(const void* p) {
  return reinterpret_cast<las_ptr>((uint32_t)reinterpret_cast<uintptr_t>(p));
}
__device__ __forceinline__ gas4_ptr to_gas4(const void* p) {
  return reinterpret_cast<gas4_ptr>(reinterpret_cast<uintptr_t>(p));
}
__device__ __forceinline__ las4_ptr to_las4(const void* p) {
  return reinterpret_cast<las4_ptr>((uint32_t)reinterpret_cast<uintptr_t>(p));
}
#endif

// Issue one 16-byte async global->LDS copy (b128 if available, else 4x b32,
// else synchronous fallback). g and l must be 16-byte aligned.
__device__ __forceinline__ void async_copy_f32x4(const float* g, float* l) {
#if HAVE_ASYNC_B128
  __builtin_amdgcn_global_load_async_to_lds_b128(to_gas4(g), to_las4(l), 0, 0);
#elif HAVE_ASYNC_LDS
  __builtin_amdgcn_global_load_async_to_lds_b32(to_gas(g + 0), to_las(l + 0), 0, 0);
  __builtin_amdgcn_global_load_async_to_lds_b32(to_gas(g + 1), to_las(l + 1), 0, 0);
  __builtin_amdgcn_global_load_async_to_lds_b32(to_gas(g + 2), to_las(l + 2), 0, 0);
  __builtin_amdgcn_global_load_async_to_lds_b32(to_gas(g + 3), to_las(l + 3), 0, 0);
#else
  l[0] = g[0]; l[1] = g[1]; l[2] = g[2]; l[3] = g[3];
#endif
}

template <int N>
__device__ __forceinline__ void wait_async() {
#if HAVE_ASYNC_LDS
  __builtin_amdgcn_s_wait_asynccnt(N);
#endif
}

// ---------------------------------------------------------------------------
// Kernel 1: per-(t,b) row logsumexp -> denom[t*B+b] = max + log(sum(exp(x-max)))
// One block per row; register-direct float4 streaming (b128 vmem), values kept
// in VGPRs across both passes; wave32 shuffle + LDS block reduction.
// ---------------------------------------------------------------------------
__global__ __launch_bounds__(256) void k_row_lse(const float* __restrict__ logits,
                                                 float* __restrict__ denom) {
  __shared__ float sred[9];
  const int row = blockIdx.x;           // row = t*BB + b
  const int tid = threadIdx.x;
  const bool act = (tid < NV4);         // 250 active threads
  const float4* __restrict__ rp4 =
      reinterpret_cast<const float4*>(logits + (size_t)row * VV);

  float4 v;
  v.x = v.y = v.z = v.w = -3.0e38f;
  if (act) v = rp4[tid];

  // max over row
  float m = fmaxf(fmaxf(v.x, v.y), fmaxf(v.z, v.w));
  for (int off = 16; off; off >>= 1) m = fmaxf(m, __shfl_xor(m, off, 32));
  if ((tid & 31) == 0) sred[tid >> 5] = m;
  __syncthreads();
  if (tid == 0) {
    float mm = sred[0];
    for (int w = 1; w < 8; ++w) mm = fmaxf(mm, sred[w]);
    sred[8] = mm;
  }
  __syncthreads();
  m = sred[8];

  // sum exp(x - max) from registers
  float s = 0.0f;
  if (act) {
    s = __expf(v.x - m) + __expf(v.y - m) + __expf(v.z - m) + __expf(v.w - m);
  }
  for (int off = 16; off; off >>= 1) s += __shfl_xor(s, off, 32);
  if ((tid & 31) == 0) sred[tid >> 5] = s;
  __syncthreads();
  if (tid == 0) {
    float ss = 0.0f;
    for (int w = 0; w < 8; ++w) ss += sred[w];
    denom[row] = m + __logf(ss);
  }
}

// ---------------------------------------------------------------------------
// Kernel 2: CTC forward recursion per batch element.
// One block per b. Alpha (L=401) ping-pongs in LDS; logits rows for t+1..t+3
// stream into a 4-deep LDS ring via async b128 loads (3-deep prefetch
// pipeline). Threads 0..249 issue exactly ROW_CNT async instructions per row,
// so constant-immediate s_wait_asynccnt works: <=2*ROW_CNT outstanding means
// row t is complete (async loads complete in order). Waves that issue nothing
// pass their wait trivially and rely on the block barrier for visibility.
// ---------------------------------------------------------------------------
__global__ __launch_bounds__(NTB) void k_ctc_forward(
    const float* __restrict__ logits, const int* __restrict__ targets,
    const int* __restrict__ in_len, const int* __restrict__ tg_len,
    const float* __restrict__ denom, float* __restrict__ partial) {
  __shared__ __align__(16) float srow[4][VV];  // ring of logits rows (4000 B/slot)
  __shared__ float salpha[2][LL + 2];          // [0..1] = NEG guard, state s at s+2
  __shared__ int   sext[LL];
  __shared__ float sskip[LL];

  const int b = blockIdx.x;
  const int tid = threadIdx.x;
  const float* __restrict__ lb = logits + (size_t)b * VV;  // row(t) = lb + t*BB*VV

  // extended labels + skip mask
  for (int l = tid; l < LL; l += NTB) {
    int lab = (l & 1) ? targets[b * SS + (l >> 1)] : 0;
    sext[l] = lab;
    float allow = 0.0f;
    if ((l & 1) && l >= 3) {
      int prev = targets[b * SS + (l >> 1) - 1];
      allow = (lab != prev) ? 1.0f : 0.0f;
    }
    sskip[l] = allow;
  }
  if (tid < 2) { salpha[0][tid] = NEGF; salpha[1][tid] = NEGF; }

  // row 0 (sync) + prefetch rows 1..3 (async) into ring slots 1..3
  for (int i = tid; i < VV; i += NTB) srow[0][i] = lb[i];
  if (tid < NV4) {
    async_copy_f32x4(lb + (size_t)1 * BB * VV + 4 * tid, &srow[1][4 * tid]);
    async_copy_f32x4(lb + (size_t)2 * BB * VV + 4 * tid, &srow[2][4 * tid]);
    async_copy_f32x4(lb + (size_t)3 * BB * VV + 4 * tid, &srow[3][4 * tid]);
  }
  __syncthreads();

  // alpha at t=0
  const float dn0 = denom[b];
  for (int l = tid; l < LL; l += NTB) {
    float a = NEGF;
    if (l == 0) a = srow[0][0] - dn0;
    else if (l == 1) a = srow[0][sext[1]] - dn0;
    salpha[0][l + 2] = a;
  }

  int cur = 0;
  const int Tlen = in_len[b];

  for (int t = 1; t < TT; ++t) {
    // wait until row t's async loads have landed (in-order completion)
    if (t < TT - 2)       wait_async<2 * ROW_CNT>();
    else if (t == TT - 2) wait_async<1 * ROW_CNT>();
    else                  wait_async<0>();
    __syncthreads();  // cross-wave visibility of row t + alpha[t-1] writes

    // issue prefetch of row t+3 into the slot last read at step t-1
    if ((t + 3 < TT) && (tid < NV4)) {
      async_copy_f32x4(lb + (size_t)(t + 3) * BB * VV + 4 * tid,
                       &srow[(t + 3) & 3][4 * tid]);
    }
    if (t + 8 < TT) __builtin_prefetch(&denom[(size_t)(t + 8) * BB + b], 0, 0);

    const float* rowp = srow[t & 3];
    const float dn = denom[(size_t)t * BB + b];   // block-uniform -> scalar load
    const int nxt = cur ^ 1;
    const bool valid = (t < Tlen);

    for (int l = tid; l < LL; l += NTB) {
      float a0 = salpha[cur][l + 2];
      float out;
      if (valid) {
        float a1 = salpha[cur][l + 1];
        float a2 = (sskip[l] != 0.0f) ? salpha[cur][l] : NEGF;
        float m = fmaxf(a0, fmaxf(a1, a2));
        float e = __expf(a0 - m) + __expf(a1 - m) + __expf(a2 - m);
        float lp = rowp[sext[l]] - dn;
        out = m + __logf(e) + lp;
      } else {
        out = a0;
      }
      salpha[nxt][l + 2] = out;
    }
    cur = nxt;
    // barrier at top of next iteration separates this write from next read
  }
  __syncthreads();

  if (tid == 0) {
    int tl = tg_len[b];
    int end = 2 * tl;
    float a_last = salpha[cur][end + 2];
    float a_prev = salpha[cur][end + 1];
    float m = fmaxf(a_last, a_prev);
    float loss = -(m + __logf(__expf(a_last - m) + __expf(a_prev - m)));
    partial[b] = loss / (float)tl;
  }
}

// ---------------------------------------------------------------------------
// Kernel 3: batch mean (32 values -> scalar)
// ---------------------------------------------------------------------------
__global__ void k_final(const float* __restrict__ partial, float* __restrict__ out) {
  float v = (threadIdx.x < BB) ? partial[threadIdx.x] : 0.0f;
  for (int off = 16; off; off >>= 1) v += __shfl_xor(v, off, 32);
  if (threadIdx.x == 0) out[0] = v / (float)BB;
}

// ---------------------------------------------------------------------------
extern "C" void kernel_launch(void* const* d_in, const int* in_sizes, int n_in,
                              void* d_out, int out_size, void* d_ws, size_t ws_size,
                              hipStream_t stream) {
  const float* logits = (const float*)d_in[0];       // [T,B,V] f32
  const int*   targets = (const int*)d_in[1];        // [B,S]
  const int*   input_lengths = (const int*)d_in[2];  // [B]
  const int*   target_lengths = (const int*)d_in[3]; // [B]

  float* denom = (float*)d_ws;                       // [T*B] = 256 KB
  float* partial = denom + (size_t)TT * BB;          // [B]

  k_row_lse<<<TT * BB, 256, 0, stream>>>(logits, denom);
  k_ctc_forward<<<BB, NTB, 0, stream>>>(logits, targets, input_lengths,
                                        target_lengths, denom, partial);
  k_final<<<1, 32, 0, stream>>>(partial, (float*)d_out);
}